// RNN_65395172049319
// MI455X (gfx1250) — compile-verified
//
#include <hip/hip_runtime.h>
#include <hip/hip_bf16.h>

// RNN: h_t = tanh(x_t @ Wx^T + b + h_{t-1} @ Wh^T)
// T=512, B=256, D=128, H=512.  fp32 matrix math via V_WMMA_F32_16X16X4_F32.
//
// Phase 1: xproj GEMM (131072x128 @ 128x512) -> written straight into d_out.
// Phase 2: persistent kernel, 64 blocks x 8 waves = 512 waves = 512 16x16 tiles
//          per step; Wh slice cached (async-DMA'd) in LDS once; grid barrier
//          between steps via device-scope atomics.

typedef __attribute__((ext_vector_type(2))) float v2f;
typedef __attribute__((ext_vector_type(8))) float v8f;

#define T_   512
#define B_   256
#define D_   128
#define H_   512
#define WROW (D_ + H_)   // 640: row stride of W (H x (D+H))
#define NB2  64          // blocks in the persistent recurrent kernel

__device__ __forceinline__ float tanh_fast(float x) {
  // tanh(x) = sign(x) * (1 - e^{-2|x|}) / (1 + e^{-2|x|}); stable for all x.
  const float NEG2LOG2E = -2.0f * 1.4426950408889634f;
  float ax = __builtin_fabsf(x);
  float e  = __builtin_amdgcn_exp2f(NEG2LOG2E * ax);   // v_exp_f32
  float r  = (1.0f - e) * __builtin_amdgcn_rcpf(1.0f + e);
  return __builtin_copysignf(r, x);
}

// WMMA f32 16x16x4. Layouts per CDNA5 ISA 7.12.2:
//   A (16x4): lane l<16 -> M=l, V0=K0,V1=K1; lanes 16..31 -> K2,K3
//   B (4x16): lane l<16 -> N=l, V0=B[K0][N],V1=B[K1][N]; lanes>=16 -> K2,K3
//   C/D     : VGPR r -> M=r (lanes 0..15), M=r+8 (lanes 16..31); N=lane%16
__device__ __forceinline__ v8f wmma_f32_k4(v2f a, v2f b, v8f c) {
  return __builtin_amdgcn_wmma_f32_16x16x4_f32(
      false, a, false, b, (short)0, c, false, false);
}

// Per-lane async global->LDS DMA (ASYNCcnt-tracked, no VGPR data traffic).
__device__ __forceinline__ void async_g2lds_b32(const float* g, const float* lds) {
  unsigned lds_off = (unsigned)(uintptr_t)lds;       // LDS byte offset
  asm volatile("global_load_async_to_lds_b32 %0, %1, off"
               :: "v"(lds_off), "v"(g) : "memory");
}
__device__ __forceinline__ void wait_async0() {
  asm volatile("s_wait_asynccnt 0x0" ::: "memory");
}

// ---------------------------------------------------------------------------
// Phase 1: out[t*B+b, h] = sum_d x[t*B+b, d] * W[h, d] + bias[h]
// Block = 8 waves; one 16-row m-tile async-staged transposed into LDS,
// each wave owns one 16-col h-tile (4 blocks cover the 32 h-tiles per m-tile).
// ---------------------------------------------------------------------------
__global__ __launch_bounds__(256) void rnn_xproj_kernel(
    const float* __restrict__ x, const float* __restrict__ W,
    const float* __restrict__ bias, float* __restrict__ out) {
  __shared__ float AT[D_ * 16];                  // AT[k*16 + r] = x[m0+r][k], 8KB
  const int tid  = threadIdx.x;
  const int lane = tid & 31;
  const int wave = tid >> 5;
  const int mtile = blockIdx.x >> 2;             // 0..8191
  const int quad  = blockIdx.x & 3;              // 0..3
  const int m0 = mtile * 16;
  const int h0 = (quad * 8 + wave) * 16;

  // Async transposed scatter: coalesced global reads, direct LDS deposit.
  for (int idx = tid; idx < 16 * D_; idx += 256) {
    int r = idx >> 7;
    int k = idx & (D_ - 1);
    async_g2lds_b32(&x[(size_t)(m0 + r) * D_ + k], &AT[k * 16 + r]);
  }
  wait_async0();
  __syncthreads();

  const int l16   = lane & 15;
  const int khalf = (lane >> 4) << 1;            // 0 or 2
  const int mhalf = (lane >> 4) << 3;            // 0 or 8

  v8f acc0 = {}, acc1 = {};                      // 2 chains -> XDL pipe overlap
  #pragma unroll 4
  for (int k = 0; k < D_; k += 8) {
    {
      const int kk = k + khalf;
      v2f a; a.x = AT[kk * 16 + l16]; a.y = AT[(kk + 1) * 16 + l16];
      float2 wv = *(const float2*)&W[(size_t)(h0 + l16) * WROW + kk];
      v2f b; b.x = wv.x; b.y = wv.y;
      acc0 = wmma_f32_k4(a, b, acc0);
    }
    {
      const int kk = k + 4 + khalf;
      v2f a; a.x = AT[kk * 16 + l16]; a.y = AT[(kk + 1) * 16 + l16];
      float2 wv = *(const float2*)&W[(size_t)(h0 + l16) * WROW + kk];
      v2f b; b.x = wv.x; b.y = wv.y;
      acc1 = wmma_f32_k4(a, b, acc1);
    }
  }
  v8f acc = acc0 + acc1;

  const int   col = h0 + l16;
  const float bv  = bias[col];
  #pragma unroll
  for (int r = 0; r < 8; ++r) {
    int row = m0 + mhalf + r;
    out[(size_t)row * H_ + col] = acc[r] + bv;
  }
}

// ---------------------------------------------------------------------------
// Phase 2: persistent recurrent kernel.
// blockIdx: htile = blk & 31 (h0 = htile*16), bhalf = blk >> 5.
// wave w owns b-tile (bhalf*8 + w). LDS holds WhT[k][n] = Wh[h0+n][k] (32KB),
// async-DMA'd once, bank-conflict-free B-operand reads for all 512 steps.
// ---------------------------------------------------------------------------
__global__ __launch_bounds__(256) void rnn_scan_kernel(
    const float* __restrict__ W, const float* __restrict__ h0in,
    float* __restrict__ out, unsigned* __restrict__ sync_cnt) {
  __shared__ float WhT[H_ * 16];                 // 32 KB
  const int tid  = threadIdx.x;
  const int lane = tid & 31;
  const int wave = tid >> 5;
  const int htile = blockIdx.x & 31;
  const int bhalf = blockIdx.x >> 5;
  const int h0    = htile * 16;
  const int brow0 = (bhalf * 8 + wave) * 16;

  // Stage Wh slice transposed: WhT[k*16+n] = W[(h0+n)*WROW + D + k]
  for (int idx = tid; idx < H_ * 16; idx += 256) {
    int n = idx >> 9;                            // 0..15
    int k = idx & (H_ - 1);                      // 0..511
    async_g2lds_b32(&W[(size_t)(h0 + n) * WROW + D_ + k], &WhT[k * 16 + n]);
  }
  wait_async0();
  __syncthreads();

  const int l16   = lane & 15;
  const int khalf = (lane >> 4) << 1;            // 0 or 2
  const int mhalf = (lane >> 4) << 3;            // 0 or 8
  const int arow  = brow0 + l16;
  const int col   = h0 + l16;

  const float* hprev = h0in;
  for (int t = 0; t < T_; ++t) {
    float* orow = out + (size_t)t * (B_ * H_);

    // Hoist the xproj tile loads: latency hidden behind the whole K-loop.
    float xp[8];
    #pragma unroll
    for (int r = 0; r < 8; ++r)
      xp[r] = orow[(size_t)(brow0 + mhalf + r) * H_ + col];

    v8f acc0 = {}, acc1 = {};
    const float* aptr = hprev + (size_t)arow * H_ + khalf;
    #pragma unroll 4
    for (int k = 0; k < H_; k += 8) {
      {
        float2 av = *(const float2*)(aptr + k);
        v2f a; a.x = av.x; a.y = av.y;
        const int kk = k + khalf;
        v2f b; b.x = WhT[kk * 16 + l16]; b.y = WhT[(kk + 1) * 16 + l16];
        acc0 = wmma_f32_k4(a, b, acc0);
      }
      {
        float2 av = *(const float2*)(aptr + k + 4);
        v2f a; a.x = av.x; a.y = av.y;
        const int kk = k + 4 + khalf;
        v2f b; b.x = WhT[kk * 16 + l16]; b.y = WhT[(kk + 1) * 16 + l16];
        acc1 = wmma_f32_k4(a, b, acc1);
      }
    }
    v8f acc = acc0 + acc1;

    #pragma unroll
    for (int r = 0; r < 8; ++r) {
      int row = brow0 + mhalf + r;
      orow[(size_t)row * H_ + col] = tanh_fast(acc[r] + xp[r]);
    }
    hprev = orow;

    // ---- grid-wide barrier (skip after the last step) ----
    if (t != T_ - 1) {
      __syncthreads();
      __threadfence();                           // agent-scope release
      if (tid == 0) {
        __hip_atomic_fetch_add(sync_cnt, 1u, __ATOMIC_ACQ_REL,
                               __HIP_MEMORY_SCOPE_AGENT);
        const unsigned target = (unsigned)NB2 * (unsigned)(t + 1);
        while (__hip_atomic_load(sync_cnt, __ATOMIC_ACQUIRE,
                                 __HIP_MEMORY_SCOPE_AGENT) < target) {
          __builtin_amdgcn_s_sleep(1);
        }
      }
      __syncthreads();
    }
  }
}

__global__ void rnn_zero_sync(unsigned* c) {
  if (threadIdx.x == 0 && blockIdx.x == 0) *c = 0u;
}

extern "C" void kernel_launch(void* const* d_in, const int* in_sizes, int n_in,
                              void* d_out, int out_size, void* d_ws, size_t ws_size,
                              hipStream_t stream) {
  const float* x   = (const float*)d_in[0];   // (T, B, D)
  const float* h0i = (const float*)d_in[1];   // (B, H)
  const float* W   = (const float*)d_in[2];   // (H, D+H)
  const float* b   = (const float*)d_in[3];   // (H,)
  float*    out = (float*)d_out;              // (T, B, H)
  unsigned* cnt = (unsigned*)d_ws;

  rnn_zero_sync<<<1, 32, 0, stream>>>(cnt);

  const int mtiles = (T_ * B_) / 16;          // 8192
  rnn_xproj_kernel<<<mtiles * 4, 256, 0, stream>>>(x, W, b, out);

  rnn_scan_kernel<<<NB2, 256, 0, stream>>>(W, h0i, out, cnt);
}